// TensorFactorizationMachineModel_12378095747625
// MI455X (gfx1250) — compile-verified
//
#include <hip/hip_runtime.h>

typedef __attribute__((ext_vector_type(2))) float v2f;
typedef __attribute__((ext_vector_type(8))) float v8f;

namespace {
constexpr int kBatch = 8192;
constexpr int kFields = 39;
constexpr int kEmbed = 64;
constexpr int kNKB = 10;      // ceil(39 / 4) K-steps for 16x16x4 WMMA
constexpr int kNL = 5;        // 2 slices (W0) + 3 slices (W1)
constexpr int kBlock = 256;   // 8 waves of 32
constexpr int kWavesPerBlock = kBlock / 32;
constexpr int kGrid = 256;    // 2048 waves -> 4 batch rows per wave
}

__global__ __launch_bounds__(kBlock) void tfm_cross_kernel(
    const int* __restrict__ x,
    const float* __restrict__ embed_table,
    const float* __restrict__ linear_w,
    const float* __restrict__ linear_bias,
    const float* __restrict__ W0,
    const float* __restrict__ W1,
    float* __restrict__ out,
    int num_waves_total)
{
  // B-matrix WMMA fragments for all 5 slices x 2 j-tiles x 10 K-steps,
  // stored in exact v_wmma_f32_16x16x4_f32 B lane layout. 25,600 bytes.
  __shared__ float wb[kNL * 2 * kNKB][32][2];

  const int tid  = threadIdx.x;
  const int lane = tid & 31;
  const int wave = tid >> 5;

  // One-time per block: swizzle the tiny constant W matrices into LDS.
  // B[k][j] = W[l][j][k]. Lane layout: lanes 0-15 -> N=lane, K=kb*4+v;
  // lanes 16-31 -> N=lane-16, K=kb*4+2+v. K >= 39 padded with zeros.
  for (int e = tid; e < kNL * 2 * kNKB * 32 * 2; e += kBlock) {
    const int v    = e & 1;
    const int ln   = (e >> 1) & 31;
    const int frag = e >> 6;               // 0..99
    const int kb   = frag % kNKB;
    const int jt   = (frag / kNKB) & 1;
    const int l5   = frag / (kNKB * 2);    // 0..4
    const int N    = jt * 16 + (ln & 15);
    const int K    = kb * 4 + ((ln >> 4) << 1) + v;
    float val = 0.0f;
    if (K < kFields) {
      val = (l5 < 2) ? W0[(l5 * 32 + N) * kFields + K]
                     : W1[((l5 - 2) * 32 + N) * kFields + K];
    }
    wb[frag][ln][v] = val;
  }
  __syncthreads();

  const float bias = linear_bias[0];
  const int gwave  = blockIdx.x * kWavesPerBlock + wave;
  const int koff   = (lane >> 4) << 1;     // 0 for lanes 0-15, 2 for 16-31
  const int mrow   = lane & 15;

  // Each wave owns whole batch rows: no cross-wave reduction needed.
  for (int b = gwave; b < kBatch; b += num_waves_total) {
    const int* xb = x + b * kFields;

    // Linear term partial sums (lanes cover k = 0..38).
    float lin = linear_w[xb[lane]];
    if (lane + 32 < kFields) lin += linear_w[xb[lane + 32]];

    float total = 0.0f;

    for (int it = 0; it < 4; ++it) {
      const int M = it * 16 + mrow;        // embed-dim row of this i-tile

      // Gather A fragments: A[i][k] = embed_table[x[b][k]][i].
      // Table is L2-resident (25.6 MB << 192 MB), reused across 10 WMMAs.
      v2f a[kNKB];
      #pragma unroll
      for (int kb = 0; kb < kNKB; ++kb) {
        const int K0 = kb * 4 + koff;
        const int K1 = K0 + 1;
        const float a0 = (K0 < kFields)
            ? embed_table[(size_t)xb[K0] * kEmbed + M] : 0.0f;
        const float a1 = (K1 < kFields)
            ? embed_table[(size_t)xb[K1] * kEmbed + M] : 0.0f;
        a[kb].x = a0;
        a[kb].y = a1;
      }

      #pragma unroll
      for (int jt = 0; jt < 2; ++jt) {
        // 5 independent accumulator chains hide WMMA->WMMA hazards.
        v8f d0 = {}, d1 = {}, d2 = {}, d3 = {}, d4 = {};
        #pragma unroll
        for (int kb = 0; kb < kNKB; ++kb) {
          const v2f b0 = *(const v2f*)&wb[(0 * 2 + jt) * kNKB + kb][lane][0];
          const v2f b1 = *(const v2f*)&wb[(1 * 2 + jt) * kNKB + kb][lane][0];
          const v2f b2 = *(const v2f*)&wb[(2 * 2 + jt) * kNKB + kb][lane][0];
          const v2f b3 = *(const v2f*)&wb[(3 * 2 + jt) * kNKB + kb][lane][0];
          const v2f b4 = *(const v2f*)&wb[(4 * 2 + jt) * kNKB + kb][lane][0];
          d0 = __builtin_amdgcn_wmma_f32_16x16x4_f32(false, a[kb], false, b0, (short)0, d0, false, false);
          d1 = __builtin_amdgcn_wmma_f32_16x16x4_f32(false, a[kb], false, b1, (short)0, d1, false, false);
          d2 = __builtin_amdgcn_wmma_f32_16x16x4_f32(false, a[kb], false, b2, (short)0, d2, false, false);
          d3 = __builtin_amdgcn_wmma_f32_16x16x4_f32(false, a[kb], false, b3, (short)0, d3, false, false);
          d4 = __builtin_amdgcn_wmma_f32_16x16x4_f32(false, a[kb], false, b4, (short)0, d4, false, false);
        }
        // Product over interaction order, elementwise on identical tile layouts,
        // then lane-local sum of this lane's 8 tile entries.
        const v8f p = d0 * d1 + d2 * d3 * d4;
        #pragma unroll
        for (int c = 0; c < 8; ++c) total += p[c];
      }
    }

    // Wave32 reduction of (cross sum + linear partials).
    float red = total + lin;
    #pragma unroll
    for (int off = 16; off > 0; off >>= 1) red += __shfl_xor(red, off, 32);
    if (lane == 0) out[b] = red + bias;
  }
}

extern "C" void kernel_launch(void* const* d_in, const int* in_sizes, int n_in,
                              void* d_out, int out_size, void* d_ws, size_t ws_size,
                              hipStream_t stream) {
  (void)in_sizes; (void)n_in; (void)out_size; (void)d_ws; (void)ws_size;
  const int*   x           = (const int*)d_in[0];
  const float* embed_table = (const float*)d_in[1];
  const float* linear_w    = (const float*)d_in[2];
  const float* linear_bias = (const float*)d_in[3];
  const float* W0          = (const float*)d_in[4];
  const float* W1          = (const float*)d_in[5];
  float*       out         = (float*)d_out;

  const int num_waves_total = kGrid * kWavesPerBlock;
  tfm_cross_kernel<<<kGrid, kBlock, 0, stream>>>(
      x, embed_table, linear_w, linear_bias, W0, W1, out, num_waves_total);
}